// GCN_38628935860963
// MI455X (gfx1250) — compile-verified
//
#include <hip/hip_runtime.h>
#include <hip/hip_bf16.h>

typedef float v2f __attribute__((ext_vector_type(2)));
typedef float v8f __attribute__((ext_vector_type(8)));

// ---------------------------------------------------------------------------
// Degree / normalization
// ---------------------------------------------------------------------------
__global__ void k_deg_init(float* __restrict__ deg, int n) {
    int i = blockIdx.x * blockDim.x + threadIdx.x;
    if (i < n) deg[i] = 1.0f;               // self-loop contributes 1 to every node
}

__global__ void k_deg_accum(const int* __restrict__ col, int E, float* __restrict__ deg) {
    int e = blockIdx.x * blockDim.x + threadIdx.x;
    if (e < E) atomicAdd(&deg[col[e]], 1.0f);
}

__global__ void k_dis(float* __restrict__ deg, int n) {
    int i = blockIdx.x * blockDim.x + threadIdx.x;
    if (i < n) {
        float d = deg[i];
        deg[i] = (d > 0.0f) ? rsqrtf(fmaxf(d, 1e-12f)) : 0.0f;   // deg -> dis in place
    }
}

// int64->int32 is not needed (indices arrive as int32); this pass compacts the
// per-edge data: row/col copies + precomputed norm = dis[row]*dis[col].
__global__ void k_norm(const int* __restrict__ ei, int E,
                       const float* __restrict__ dis,
                       int* __restrict__ row32, int* __restrict__ col32,
                       float* __restrict__ nrm) {
    int e = blockIdx.x * blockDim.x + threadIdx.x;
    if (e >= E) return;
    int r = ei[e];
    int c = ei[E + e];
    row32[e] = r;
    col32[e] = c;
    nrm[e]   = dis[r] * dis[c];
}

// ---------------------------------------------------------------------------
// fp32 WMMA GEMM:  C[M x NN] = A[M x K] @ B[K x NN]   (all row-major)
// One wave computes a 16 x NN strip: NT = NN/16 accumulators (8 VGPRs each),
// A fragment loaded once per K-step and reused across all NT B tiles.
// V_WMMA_F32_16X16X4_F32 layouts (wave32):
//   A 16x4 : lane r(0-15)/r+16 holds A[r][kk], A[r][kk+1], kk = (lane>=16)?2:0
//   B 4x16 : lane n(0-15)/n+16 holds B[kk][n], B[kk+1][n]
//   C 16x16: VGPR v, lanes 0-15 -> C[v][n], lanes 16-31 -> C[v+8][n]
// ---------------------------------------------------------------------------
template <int K, int NN>
__global__ void k_gemm_wmma_f32(const float* __restrict__ A,
                                const float* __restrict__ B,
                                float* __restrict__ C, int m_tiles) {
    const int wave = blockIdx.x * (blockDim.x >> 5) + (threadIdx.x >> 5);
    if (wave >= m_tiles) return;            // wave-uniform: EXEC stays all-ones
    const int lane = threadIdx.x & 31;
    const int r    = lane & 15;             // row within A tile / col within B tile
    const int hi   = lane >> 4;             // 0 or 1
    const int kk   = hi * 2;                // K sub-offset within 4-wide step
    const int m0   = wave * 16;

    constexpr int NT = NN / 16;
    v8f acc[NT];
    const v8f vzero = {0.f, 0.f, 0.f, 0.f, 0.f, 0.f, 0.f, 0.f};
#pragma unroll
    for (int t = 0; t < NT; ++t) acc[t] = vzero;

    const float* arow = A + (size_t)(m0 + r) * K;
    for (int k = 0; k < K; k += 4) {
        // A fragment: 2 consecutive fp32 -> one B64 load
        const float2 af = *(const float2*)(arow + k + kk);
        v2f a; a.x = af.x; a.y = af.y;
#pragma unroll
        for (int t = 0; t < NT; ++t) {
            const int n = t * 16 + r;
            v2f b;
            b.x = B[(size_t)(k + kk) * NN + n];
            b.y = B[(size_t)(k + kk + 1) * NN + n];
            acc[t] = __builtin_amdgcn_wmma_f32_16x16x4_f32(
                false, a, false, b, (short)0, acc[t], false, false);
        }
    }

#pragma unroll
    for (int t = 0; t < NT; ++t) {
        const int n = t * 16 + r;
#pragma unroll
        for (int v = 0; v < 8; ++v) {
            C[(size_t)(m0 + v + hi * 8) * NN + n] = acc[t][v];
        }
    }
}

// ---------------------------------------------------------------------------
// Edge aggregation: acc[col[e]][c..c+3] += h[row[e]][c..c+3] * norm[e]
// Consecutive threads cover the feature chunks of one edge -> coalesced
// float4 gathers and contiguous scalar f32 atomics on the destination row.
// ---------------------------------------------------------------------------
template <int F>
__global__ void k_agg(const float* __restrict__ h, float* __restrict__ acc,
                      const int* __restrict__ row, const int* __restrict__ col,
                      const float* __restrict__ nrm, int E) {
    constexpr int C4 = F / 4;
    const int t = blockIdx.x * blockDim.x + threadIdx.x;   // < E*C4 (max 51.2M)
    if (t >= E * C4) return;
    const int e = t / C4;
    const int c = (t - e * C4) * 4;
    const int r  = row[e];
    const int cl = col[e];
    const float w = nrm[e];
    const float4 hv = *(const float4*)(h + (size_t)r * F + c);
    float* dst = acc + (size_t)cl * F + c;
    atomicAdd(dst + 0, hv.x * w);
    atomicAdd(dst + 1, hv.y * w);
    atomicAdd(dst + 2, hv.z * w);
    atomicAdd(dst + 3, hv.w * w);
}

// ---------------------------------------------------------------------------
// Layer-1 post: fold self-loop (h1[i]*dis[i]^2), bias, ReLU; in place in acc1.
// ---------------------------------------------------------------------------
__global__ void k_post1_relu(float* __restrict__ acc, const float* __restrict__ h,
                             const float* __restrict__ dis, const float* __restrict__ b,
                             int total) {
    int t = blockIdx.x * blockDim.x + threadIdx.x;
    if (t >= total) return;
    const int i = t >> 7;            // F = 128
    const int f = t & 127;
    const float sn = dis[i] * dis[i];
    const float v = acc[t] + h[t] * sn + b[f];
    acc[t] = v > 0.0f ? v : 0.0f;
}

// ---------------------------------------------------------------------------
// Layer-2 post + log_softmax: one wave32 per row of 64, 2 cols/lane,
// shfl_xor tree reductions for max and sum(exp). In place in d_out.
// ---------------------------------------------------------------------------
__global__ void k_post2_lsm(float* __restrict__ z, const float* __restrict__ h,
                            const float* __restrict__ dis, const float* __restrict__ b,
                            int n) {
    const int wave = blockIdx.x * (blockDim.x >> 5) + (threadIdx.x >> 5);
    if (wave >= n) return;
    const int lane = threadIdx.x & 31;
    const int i = wave;
    const int c = lane * 2;
    const float sn = dis[i] * dis[i];
    const size_t base = (size_t)i * 64 + c;
    float z0 = z[base + 0] + h[base + 0] * sn + b[c + 0];
    float z1 = z[base + 1] + h[base + 1] * sn + b[c + 1];

    float m = fmaxf(z0, z1);
#pragma unroll
    for (int off = 16; off >= 1; off >>= 1)
        m = fmaxf(m, __shfl_xor(m, off, 32));
    float s = expf(z0 - m) + expf(z1 - m);
#pragma unroll
    for (int off = 16; off >= 1; off >>= 1)
        s += __shfl_xor(s, off, 32);
    const float lse = m + logf(s);
    z[base + 0] = z0 - lse;
    z[base + 1] = z1 - lse;
}

// ---------------------------------------------------------------------------
extern "C" void kernel_launch(void* const* d_in, const int* in_sizes, int n_in,
                              void* d_out, int out_size, void* d_ws, size_t ws_size,
                              hipStream_t stream) {
    const float* x  = (const float*)d_in[0];   // [N,256]
    const int*   ei = (const int*)d_in[1];     // [2,E]
    const float* W1 = (const float*)d_in[2];   // [256,128]
    const float* b1 = (const float*)d_in[3];   // [128]
    const float* W2 = (const float*)d_in[4];   // [128,64]
    const float* b2 = (const float*)d_in[5];   // [64]
    float* out = (float*)d_out;                // [N,64]

    const int N = in_sizes[0] / 256;           // 100000
    const int E = in_sizes[1] / 2;             // 1600000

    // Workspace layout (floats), ~148 MB total:
    float* dis   = (float*)d_ws;               // N       (deg -> dis in place)
    float* nrm   = dis + N;                    // E
    int*   row32 = (int*)(nrm + E);            // E
    int*   col32 = row32 + E;                  // E
    float* h1    = (float*)(col32 + E);        // N*128
    float* acc1  = h1 + (size_t)N * 128;       // N*128
    float* h2    = acc1 + (size_t)N * 128;     // N*64

    const int tb = 256;

    // --- gcn_norm ---
    k_deg_init<<<(N + tb - 1) / tb, tb, 0, stream>>>(dis, N);
    k_deg_accum<<<(E + tb - 1) / tb, tb, 0, stream>>>(ei + E, E, dis);
    k_dis<<<(N + tb - 1) / tb, tb, 0, stream>>>(dis, N);
    k_norm<<<(E + tb - 1) / tb, tb, 0, stream>>>(ei, E, dis, row32, col32, nrm);

    const int m_tiles = N / 16;                // 6250, exact
    const int gemm_blocks = (m_tiles + 3) / 4; // 4 waves (128 thr) per block

    // --- layer 1 ---
    k_gemm_wmma_f32<256, 128><<<gemm_blocks, 128, 0, stream>>>(x, W1, h1, m_tiles);
    hipMemsetAsync(acc1, 0, (size_t)N * 128 * sizeof(float), stream);
    {
        const int tot = E * 32;                // E * 128/4
        k_agg<128><<<(tot + tb - 1) / tb, tb, 0, stream>>>(h1, acc1, row32, col32, nrm, E);
    }
    k_post1_relu<<<(N * 128 + tb - 1) / tb, tb, 0, stream>>>(acc1, h1, dis, b1, N * 128);

    // --- layer 2 ---
    k_gemm_wmma_f32<128, 64><<<gemm_blocks, 128, 0, stream>>>(acc1, W2, h2, m_tiles);
    hipMemsetAsync(out, 0, (size_t)N * 64 * sizeof(float), stream);
    {
        const int tot = E * 16;                // E * 64/4
        k_agg<64><<<(tot + tb - 1) / tb, tb, 0, stream>>>(h2, out, row32, col32, nrm, E);
    }
    k_post2_lsm<<<(N + 7) / 8, 256, 0, stream>>>(out, h2, dis, b2, N);
}